// GAT_52604759441721
// MI455X (gfx1250) — compile-verified
//
#include <hip/hip_runtime.h>

// ---- problem constants (match reference) ----
#define B_     4
#define FIN_   354
#define N_     4096
#define E_     65536
#define ETOT_  (E_ + N_)   // + self loops
#define HID_   352
#define HEADS_ 4
#define PH_    88
#define EPS_   1e-5f
#define SLOPE_ 0.2f

typedef float v2f __attribute__((ext_vector_type(2)));
typedef float v8f __attribute__((ext_vector_type(8)));

// order-preserving float<->uint encoding for atomicMax-based segment max
__device__ __forceinline__ unsigned enc_f(float f) {
    unsigned u = __float_as_uint(f);
    return u ^ ((u >> 31) ? 0xFFFFFFFFu : 0x80000000u);
}
__device__ __forceinline__ float dec_f(unsigned u) {
    unsigned b = (u & 0x80000000u) ? (u ^ 0x80000000u) : (u ^ 0xFFFFFFFFu);
    return __uint_as_float(b);
}
__device__ __forceinline__ float leakyf(float x) { return x > 0.f ? x : SLOPE_ * x; }

__device__ __forceinline__ void edge_sd(const int* __restrict__ ei, int e, int& s, int& d) {
    if (e < E_) { s = ei[e]; d = ei[E_ + e]; }
    else        { s = d = e - E_; }           // appended self loops
}

// ---------------------------------------------------------------------------
// GEMM: Y[g] = A[g] @ W   (compile-time strides: layer 1 reads x^T in place)
// wave -> 16x32 tile of Y via two V_WMMA_F32_16X16X4_F32 accumulators.
// A 16x4 f32 layout: lane l: M=l&15, VGPR0=K(2*(l>>4)), VGPR1=K(2*(l>>4)+1)
// B 4x16 f32 layout: symmetric (N=l&15).  C/D: VGPR r -> M=r (+8 for lanes>=16)
// Main loop is branch-free and software-pipelined: next fragments are loaded
// before the current WMMA pair so load latency hides under matrix-pipe work.
// ---------------------------------------------------------------------------
template<int K, int RS, int CS>
__global__ __launch_bounds__(128) void gemm_wmma_kernel(
    const float* __restrict__ A, long aGraphStride,
    const float* __restrict__ W, float* __restrict__ Y)
{
    static_assert((K % 2) == 0, "tail handling assumes even K");
    const int lane = threadIdx.x & 31;
    const int wave = threadIdx.x >> 5;
    const float* Ag = A + (long)blockIdx.z * aGraphStride;
    float*       Yg = Y + (long)blockIdx.z * N_ * HID_;

    const int rowBase = (blockIdx.x * 4 + wave) * 16;
    const int colBase = blockIdx.y * 32;
    const int m  = lane & 15;       // row within A tile / col within B tile
    const int kh = lane >> 4;       // which K-half this lane holds

    const float* aPtr = Ag + (long)(rowBase + m) * RS + (long)(2 * kh) * CS;
    const float* wPtr = W + (2 * kh) * HID_ + colBase + m;

    v8f acc0 = {};
    v8f acc1 = {};

    constexpr int K4 = K & ~3;      // full 4-wide K steps

    // prologue: fragments for k0 = 0
    v2f a, b0, b1;
    a.x  = aPtr[0];    a.y  = aPtr[CS];
    b0.x = wPtr[0];    b0.y = wPtr[HID_];
    b1.x = wPtr[16];   b1.y = wPtr[HID_ + 16];

    for (int k0 = 4; k0 < K4; k0 += 4) {
        aPtr += 4 * (long)CS;
        wPtr += 4 * HID_;
        v2f an, b0n, b1n;                     // issue next-step loads first
        an.x  = aPtr[0];   an.y  = aPtr[CS];
        b0n.x = wPtr[0];   b0n.y = wPtr[HID_];
        b1n.x = wPtr[16];  b1n.y = wPtr[HID_ + 16];

        acc0 = __builtin_amdgcn_wmma_f32_16x16x4_f32(false, a, false, b0,
                                                     (short)0, acc0, false, false);
        acc1 = __builtin_amdgcn_wmma_f32_16x16x4_f32(false, a, false, b1,
                                                     (short)0, acc1, false, false);
        a = an; b0 = b0n; b1 = b1n;
    }
    // last full step
    acc0 = __builtin_amdgcn_wmma_f32_16x16x4_f32(false, a, false, b0,
                                                 (short)0, acc0, false, false);
    acc1 = __builtin_amdgcn_wmma_f32_16x16x4_f32(false, a, false, b1,
                                                 (short)0, acc1, false, false);

    if constexpr (K4 < K) {
        // K % 4 == 2 tail: only the kh==0 half-wave holds valid K indices
        aPtr += 4 * (long)CS;
        wPtr += 4 * HID_;
        v2f at = {0.f, 0.f}, b0t = {0.f, 0.f}, b1t = {0.f, 0.f};
        if (kh == 0) {
            at.x  = aPtr[0];   at.y  = aPtr[CS];
            b0t.x = wPtr[0];   b0t.y = wPtr[HID_];
            b1t.x = wPtr[16];  b1t.y = wPtr[HID_ + 16];
        }
        acc0 = __builtin_amdgcn_wmma_f32_16x16x4_f32(false, at, false, b0t,
                                                     (short)0, acc0, false, false);
        acc1 = __builtin_amdgcn_wmma_f32_16x16x4_f32(false, at, false, b1t,
                                                     (short)0, acc1, false, false);
    }

    const int nn = lane & 15;
    const int mo = (lane >> 4) * 8;
    #pragma unroll
    for (int r = 0; r < 8; ++r) {
        const int row = rowBase + mo + r;
        Yg[(long)row * HID_ + colBase + nn]      = acc0[r];
        Yg[(long)row * HID_ + colBase + 16 + nn] = acc1[r];
    }
}

// per-(node,head) attention dot products: a_src/a_dst
__global__ void attn_kernel(const float* __restrict__ XH,
                            const float* __restrict__ as_,
                            const float* __restrict__ ad_,
                            float* __restrict__ ASRC, float* __restrict__ ADST)
{
    int idx = blockIdx.x * blockDim.x + threadIdx.x;
    if (idx >= B_ * N_ * HEADS_) return;
    int h = idx % HEADS_;
    int n = (idx / HEADS_) % N_;
    int b = idx / (N_ * HEADS_);
    const float* xrow = XH + ((long)b * N_ + n) * HID_ + h * PH_;
    float ss = 0.f, sd = 0.f;
    for (int c = 0; c < PH_; ++c) {
        float v = xrow[c];
        ss += v * as_[h * PH_ + c];
        sd += v * ad_[h * PH_ + c];
    }
    ASRC[idx] = ss;
    ADST[idx] = sd;
}

__global__ void init_agg_kernel(float* __restrict__ AGG, const float* __restrict__ bias)
{
    long idx = (long)blockIdx.x * blockDim.x + threadIdx.x;
    if (idx >= (long)B_ * N_ * HID_) return;
    AGG[idx] = bias[idx % HID_];
}

__global__ void init_small_kernel(unsigned* __restrict__ MENC, float* __restrict__ DEN)
{
    int idx = blockIdx.x * blockDim.x + threadIdx.x;
    if (idx >= B_ * N_ * HEADS_) return;
    MENC[idx] = 0u;      // encoded "-infinity" (every node has a self loop)
    DEN[idx]  = 0.f;
}

__global__ void edge_max_kernel(const int* __restrict__ ei,
                                const float* __restrict__ ASRC,
                                const float* __restrict__ ADST,
                                unsigned* __restrict__ MENC)
{
    long idx = (long)blockIdx.x * blockDim.x + threadIdx.x;
    if (idx >= (long)B_ * ETOT_ * HEADS_) return;
    int h = (int)(idx % HEADS_);
    int e = (int)((idx / HEADS_) % ETOT_);
    int b = (int)(idx / ((long)ETOT_ * HEADS_));
    int s, d; edge_sd(ei, e, s, d);
    float lg = leakyf(ASRC[((long)b * N_ + s) * HEADS_ + h] +
                      ADST[((long)b * N_ + d) * HEADS_ + h]);
    atomicMax(&MENC[((long)b * N_ + d) * HEADS_ + h], enc_f(lg));
}

__global__ void edge_exp_kernel(const int* __restrict__ ei,
                                const float* __restrict__ ASRC,
                                const float* __restrict__ ADST,
                                const unsigned* __restrict__ MENC,
                                float* __restrict__ DEN, float* __restrict__ EX)
{
    long idx = (long)blockIdx.x * blockDim.x + threadIdx.x;
    if (idx >= (long)B_ * ETOT_ * HEADS_) return;
    int h = (int)(idx % HEADS_);
    int e = (int)((idx / HEADS_) % ETOT_);
    int b = (int)(idx / ((long)ETOT_ * HEADS_));
    int s, d; edge_sd(ei, e, s, d);
    float lg = leakyf(ASRC[((long)b * N_ + s) * HEADS_ + h] +
                      ADST[((long)b * N_ + d) * HEADS_ + h]);
    float m  = dec_f(MENC[((long)b * N_ + d) * HEADS_ + h]);
    float ex = __expf(lg - m);
    EX[idx] = ex;
    atomicAdd(&DEN[((long)b * N_ + d) * HEADS_ + h], ex);
}

__global__ void edge_alpha_kernel(const int* __restrict__ ei,
                                  const float* __restrict__ DEN,
                                  float* __restrict__ EX)
{
    long idx = (long)blockIdx.x * blockDim.x + threadIdx.x;
    if (idx >= (long)B_ * ETOT_ * HEADS_) return;
    int h = (int)(idx % HEADS_);
    int e = (int)((idx / HEADS_) % ETOT_);
    int b = (int)(idx / ((long)ETOT_ * HEADS_));
    int s, d; edge_sd(ei, e, s, d);
    EX[idx] = EX[idx] / DEN[((long)b * N_ + d) * HEADS_ + h];
}

__global__ void edge_aggr_kernel(const int* __restrict__ ei,
                                 const float* __restrict__ EX,
                                 const float* __restrict__ XH,
                                 float* __restrict__ AGG)
{
    long idx = (long)blockIdx.x * blockDim.x + threadIdx.x;
    if (idx >= (long)B_ * ETOT_ * HID_) return;
    int c = (int)(idx % HID_);
    int e = (int)((idx / HID_) % ETOT_);
    int b = (int)(idx / ((long)ETOT_ * HID_));
    int s, d; edge_sd(ei, e, s, d);
    float alpha = EX[((long)b * ETOT_ + e) * HEADS_ + (c / PH_)];
    atomicAdd(&AGG[((long)b * N_ + d) * HID_ + c],
              alpha * XH[((long)b * N_ + s) * HID_ + c]);
}

// GraphNorm (+optional residual) + ReLU, fused.  32 features x 8 node-stripes.
__global__ void gnorm_kernel(const float* __restrict__ AGG,
                             const float* __restrict__ HPREV,
                             const float* __restrict__ w,
                             const float* __restrict__ bb,
                             const float* __restrict__ ms,
                             float* __restrict__ HOUT, int residual)
{
    __shared__ float s1[8][32], s2[8][32];
    __shared__ float mu_s[32], inv_s[32], ms_s[32];
    const int tx = threadIdx.x, ty = threadIdx.y;
    const int c  = blockIdx.x * 32 + tx;
    const int b  = blockIdx.y;
    const float* X = AGG + (long)b * N_ * HID_;

    float sum = 0.f, sq = 0.f;
    for (int n = ty; n < N_; n += 8) {
        float v = X[(long)n * HID_ + c];
        sum += v; sq += v * v;
    }
    s1[ty][tx] = sum; s2[ty][tx] = sq;
    __syncthreads();
    if (ty == 0) {
        float S = 0.f, Q = 0.f;
        #pragma unroll
        for (int i = 0; i < 8; ++i) { S += s1[i][tx]; Q += s2[i][tx]; }
        float mu  = S / (float)N_;
        float msv = ms[c];
        float var = Q / (float)N_ - (2.f * msv - msv * msv) * mu * mu;
        mu_s[tx] = mu; ms_s[tx] = msv; inv_s[tx] = rsqrtf(var + EPS_);
    }
    __syncthreads();
    const float mu = mu_s[tx], inv = inv_s[tx], msv = ms_s[tx];
    const float wc = w[c], bc = bb[c];
    const float* P = residual ? (HPREV + (long)b * N_ * HID_) : nullptr;
    float* O = HOUT + (long)b * N_ * HID_;
    for (int n = ty; n < N_; n += 8) {
        float v = X[(long)n * HID_ + c];
        float y = (v - msv * mu) * inv * wc + bc;
        if (residual) y += P[(long)n * HID_ + c];
        O[(long)n * HID_ + c] = fmaxf(y, 0.f);
    }
}

// global mean pool + Linear(352 -> 1); one block per graph
__global__ void pool_kernel(const float* __restrict__ H,
                            const float* __restrict__ cw,
                            const float* __restrict__ cb,
                            float* __restrict__ out)
{
    __shared__ float red[256];
    const int b = blockIdx.x, t = threadIdx.x;
    const float* X = H + (long)b * N_ * HID_;
    float acc = 0.f;
    for (int c = t; c < HID_; c += 256) {
        float s = 0.f;
        for (int n = 0; n < N_; ++n) s += X[(long)n * HID_ + c];
        acc += (s / (float)N_) * cw[c];
    }
    red[t] = acc;
    __syncthreads();
    for (int o = 128; o > 0; o >>= 1) {
        if (t < o) red[t] += red[t + o];
        __syncthreads();
    }
    if (t == 0) out[b] = red[0] + cb[0];
}

// ---------------------------------------------------------------------------
extern "C" void kernel_launch(void* const* d_in, const int* in_sizes, int n_in,
                              void* d_out, int out_size, void* d_ws, size_t ws_size,
                              hipStream_t stream)
{
    (void)in_sizes; (void)n_in; (void)out_size; (void)ws_size;

    const float* x  = (const float*)d_in[0];
    const int*   ei = (const int*)d_in[1];
    const float* W_[3]  = {(const float*)d_in[2],  (const float*)d_in[9],  (const float*)d_in[16]};
    const float* AS_[3] = {(const float*)d_in[3],  (const float*)d_in[10], (const float*)d_in[17]};
    const float* AD_[3] = {(const float*)d_in[4],  (const float*)d_in[11], (const float*)d_in[18]};
    const float* BI_[3] = {(const float*)d_in[5],  (const float*)d_in[12], (const float*)d_in[19]};
    const float* GW_[3] = {(const float*)d_in[6],  (const float*)d_in[13], (const float*)d_in[20]};
    const float* GB_[3] = {(const float*)d_in[7],  (const float*)d_in[14], (const float*)d_in[21]};
    const float* GM_[3] = {(const float*)d_in[8],  (const float*)d_in[15], (const float*)d_in[22]};
    const float* cw = (const float*)d_in[23];
    const float* cb = (const float*)d_in[24];
    float* out = (float*)d_out;

    // workspace carve-up
    char* ws = (char*)d_ws;
    const size_t SZ  = (size_t)B_ * N_ * HID_ * sizeof(float);       // 23.07 MB
    const size_t SZE = (size_t)B_ * ETOT_ * HEADS_ * sizeof(float);  // 4.46 MB
    const size_t SZS = (size_t)B_ * N_ * HEADS_ * sizeof(float);     // 256 KB
    float*    XH   = (float*)(ws);
    float*    AGG  = (float*)(ws + SZ);
    float*    H1   = (float*)(ws + 2 * SZ);
    float*    H2   = (float*)(ws + 3 * SZ);
    float*    EX   = (float*)(ws + 4 * SZ);
    float*    ASRC = (float*)(ws + 4 * SZ + SZE);
    float*    ADST = (float*)((char*)ASRC + SZS);
    unsigned* MENC = (unsigned*)((char*)ADST + SZS);
    float*    DEN  = (float*)((char*)MENC + SZS);

    const float* hin[3]  = {x, H1, H2};
    float*       hout[3] = {H1, H2, H1};

    const int  nNH = B_ * N_ * HEADS_;
    const long nAG = (long)B_ * N_ * HID_;
    const long nEH = (long)B_ * ETOT_ * HEADS_;
    const long nEC = (long)B_ * ETOT_ * HID_;

    const dim3 gemmGrid(N_ / 64, HID_ / 32, B_);

    for (int l = 0; l < 3; ++l) {
        if (l == 0) {
            // layer 1: A = x^T in place (row stride 1, col stride N), K = FIN
            gemm_wmma_kernel<FIN_, 1, N_><<<gemmGrid, 128, 0, stream>>>(
                x, (long)FIN_ * N_, W_[0], XH);
        } else {
            // layers 2/3: row-major activations, K = HID
            gemm_wmma_kernel<HID_, HID_, 1><<<gemmGrid, 128, 0, stream>>>(
                hin[l], (long)N_ * HID_, W_[l], XH);
        }

        attn_kernel<<<(nNH + 255) / 256, 256, 0, stream>>>(XH, AS_[l], AD_[l], ASRC, ADST);

        init_agg_kernel<<<(unsigned)((nAG + 255) / 256), 256, 0, stream>>>(AGG, BI_[l]);
        init_small_kernel<<<(nNH + 255) / 256, 256, 0, stream>>>(MENC, DEN);

        edge_max_kernel<<<(unsigned)((nEH + 255) / 256), 256, 0, stream>>>(ei, ASRC, ADST, MENC);
        edge_exp_kernel<<<(unsigned)((nEH + 255) / 256), 256, 0, stream>>>(ei, ASRC, ADST, MENC, DEN, EX);
        edge_alpha_kernel<<<(unsigned)((nEH + 255) / 256), 256, 0, stream>>>(ei, DEN, EX);
        edge_aggr_kernel<<<(unsigned)((nEC + 255) / 256), 256, 0, stream>>>(ei, EX, XH, AGG);

        gnorm_kernel<<<dim3(HID_ / 32, B_), dim3(32, 8), 0, stream>>>(
            AGG, l == 0 ? (const float*)nullptr : hin[l],
            GW_[l], GB_[l], GM_[l], hout[l], l == 0 ? 0 : 1);
    }

    pool_kernel<<<B_, 256, 0, stream>>>(H1, cw, cb, out);
}